// GCN_29978871726611
// MI455X (gfx1250) — compile-verified
//
#include <hip/hip_runtime.h>
#include <hip/hip_bf16.h>
#include <math.h>

// ---------------- problem constants (match reference) ----------------
#define N_NODES 50000
#define N_EDGES 800000
#define HID     128
#define OUTC    40
#define NLAYERS 3
#define BN_EPS  1e-5f
#define WT_STRIDE 130   // padded LDS stride (floats): even -> 8B-aligned float2, conflict-free

typedef __attribute__((ext_vector_type(2))) float v2f;
typedef __attribute__((ext_vector_type(8))) float v8f;

__device__ __forceinline__ void atomAddF32(float* p, float v) {
    __hip_atomic_fetch_add(p, v, __ATOMIC_RELAXED, __HIP_MEMORY_SCOPE_AGENT);
}

// ---------------- degree / normalization ----------------
__global__ void gcn_init_deg(float* dinv, int n) {
    int i = blockIdx.x * blockDim.x + threadIdx.x;
    if (i < n) dinv[i] = 1.0f;   // self-loop
}

__global__ void gcn_deg_edges(const int* __restrict__ ei, float* dinv, int e) {
    int i = blockIdx.x * blockDim.x + threadIdx.x;
    if (i < e) atomAddF32(&dinv[ei[e + i]], 1.0f);   // dst row
}

__global__ void gcn_finish_dinv(float* dinv, int n) {
    int i = blockIdx.x * blockDim.x + threadIdx.x;
    if (i < n) dinv[i] = rsqrtf(dinv[i]);
}

// ---------------- h0 = alpha*x[:, :128] + beta*x[:, 128:] ----------------
__global__ void gcn_combine(const float* __restrict__ x,
                            const float* __restrict__ la,
                            const float* __restrict__ lb,
                            float* __restrict__ h, int n) {
    int idx = blockIdx.x * blockDim.x + threadIdx.x;
    if (idx >= n * HID) return;
    int nd = idx >> 7, f = idx & 127;
    float alpha = expf(la[0]);
    float beta  = expf(lb[0]);
    h[idx] = alpha * x[(size_t)nd * 256 + f] + beta * x[(size_t)nd * 256 + 128 + f];
}

// ---------------- WMMA fp32 GEMM: C[nrows x 128] = A[nrows x 128] * W[128 x 128] ----------------
// block = 128 threads (4 waves). Each wave computes a 16x128 strip of C with
// 8 persistent 16x16 f32 accumulators; 32 K-steps of V_WMMA_F32_16X16X4_F32.
// Weight tile staged TRANSPOSED (Wt[n][k], padded stride) so each B fragment is
// one contiguous 8-byte ds_load_b64 straight into an even-aligned VGPR pair.
__global__ __launch_bounds__(128) void gcn_gemm128_wmma(
        const float* __restrict__ A, const float* __restrict__ W,
        const float* __restrict__ bias, float* __restrict__ C,
        int nrows, int do_relu) {
    __shared__ float Wt[HID * WT_STRIDE];   // ~65 KB (CDNA5 WGP: up to 320 KB)

    int t = threadIdx.x;
    // cooperative transposed load: Wt[n*WT_STRIDE + k] = W[k*128 + n]
    // per sweep j: threads read W[j*128 + t] (coalesced), write banks (2t+j)%64 (conflict-free)
    #pragma unroll 4
    for (int idx = t; idx < HID * HID; idx += 128) {
        int kk = idx >> 7;
        int n  = idx & 127;
        Wt[n * WT_STRIDE + kk] = W[idx];
    }
    __syncthreads();

    const int wave  = t >> 5;
    const int lane  = t & 31;
    const int row16 = lane & 15;          // M (for A) / N (for B) within tile
    const int khalf = (lane >> 4) << 1;   // 0 for lanes 0-15, 2 for lanes 16-31

    const int tileM = blockIdx.x * 64 + wave * 16;
    int rowIdx = tileM + row16;
    if (rowIdx > nrows - 1) rowIdx = nrows - 1;       // clamp (stores masked later)
    const float* Arow = A + (size_t)rowIdx * HID;
    const float* Brow = Wt + (size_t)row16 * WT_STRIDE;  // this lane's N column, all K

    v8f acc[8] = {};

    for (int k = 0; k < HID; k += 4) {
        // A fragment: 16x4, V0 = K={khalf}, V1 = K={khalf+1}
        const float2 av = *(const float2*)(Arow + k + khalf);
        v2f a; a.x = av.x; a.y = av.y;
        #pragma unroll
        for (int nt = 0; nt < 8; ++nt) {
            // B fragment: contiguous in transposed LDS -> single ds_load_b64
            const float2 bv = *(const float2*)(Brow + nt * 16 * WT_STRIDE + k + khalf);
            v2f b; b.x = bv.x; b.y = bv.y;
            acc[nt] = __builtin_amdgcn_wmma_f32_16x16x4_f32(
                false, a, false, b, (short)0, acc[nt], false, false);
        }
    }

    // epilogue: C layout — VGPR r: lanes 0-15 -> M=r, lanes 16-31 -> M=r+8
    const int rbase = tileM + ((lane >> 4) << 3);
    #pragma unroll
    for (int nt = 0; nt < 8; ++nt) {
        const int col = nt * 16 + row16;
        const float bv = bias ? bias[col] : 0.0f;
        #pragma unroll
        for (int r = 0; r < 8; ++r) {
            const int row = rbase + r;
            if (row < nrows) {
                float v = acc[nt][r] + bv;
                if (do_relu) v = fmaxf(v, 0.0f);
                C[(size_t)row * HID + col] = v;
            }
        }
    }
}

// ---------------- self-loop + conv bias: h[n,f] = b[f] + m[n,f]*dinv[n]^2 ----------------
__global__ void gcn_selfloop(const float* __restrict__ m, const float* __restrict__ dinv,
                             const float* __restrict__ b, float* __restrict__ h, int n) {
    int idx = blockIdx.x * blockDim.x + threadIdx.x;
    if (idx >= n * HID) return;
    int nd = idx >> 7, f = idx & 127;
    float di = dinv[nd];
    h[idx] = b[f] + m[idx] * di * di;
}

// ---------------- edge scatter: h[dst] += m[src] * dinv[src]*dinv[dst] ----------------
// one wave32 per edge: lane handles 4 contiguous features (float4 gather + 4 fp32 atomics)
__global__ __launch_bounds__(256) void gcn_scatter(const int* __restrict__ ei,
                                                   const float* __restrict__ dinv,
                                                   const float* __restrict__ m,
                                                   float* __restrict__ h, int e) {
    int w = (blockIdx.x * blockDim.x + threadIdx.x) >> 5;
    int lane = threadIdx.x & 31;
    if (w >= e) return;
    int s = ei[w];
    int d = ei[e + w];
    float nrm = dinv[s] * dinv[d];
    const float4 v = *(const float4*)(m + (size_t)s * HID + lane * 4);
    float* hp = h + (size_t)d * HID + lane * 4;
    atomAddF32(hp + 0, v.x * nrm);
    atomAddF32(hp + 1, v.y * nrm);
    atomAddF32(hp + 2, v.z * nrm);
    atomAddF32(hp + 3, v.w * nrm);
}

// ---------------- batch norm ----------------
__global__ void gcn_bn_zero(float* bnsum, float* bnsq) {
    int t = threadIdx.x;
    if (t < HID) bnsum[t] = 0.0f;
    else if (t < 2 * HID) bnsq[t - HID] = 0.0f;
}

__global__ __launch_bounds__(128) void gcn_bn_stats(const float* __restrict__ h,
                                                    float* bnsum, float* bnsq, int n) {
    int f = threadIdx.x;                 // one feature per thread
    int chunk = (n + gridDim.x - 1) / gridDim.x;
    int n0 = blockIdx.x * chunk;
    int n1 = n0 + chunk; if (n1 > n) n1 = n;
    float s = 0.0f, ss = 0.0f;
    for (int nd = n0; nd < n1; ++nd) {
        float v = h[(size_t)nd * HID + f];
        s += v; ss += v * v;
    }
    atomAddF32(&bnsum[f], s);
    atomAddF32(&bnsq[f], ss);
}

__global__ void gcn_bn_apply(float* __restrict__ h,
                             const float* __restrict__ bnsum, const float* __restrict__ bnsq,
                             const float* __restrict__ gamma, const float* __restrict__ beta,
                             int n) {
    int idx = blockIdx.x * blockDim.x + threadIdx.x;
    if (idx >= n * HID) return;
    int f = idx & 127;
    float invn = 1.0f / (float)n;
    float mu  = bnsum[f] * invn;
    float var = bnsq[f] * invn - mu * mu;
    float v = gamma[f] * (h[idx] - mu) * rsqrtf(var + BN_EPS) + beta[f];
    h[idx] = fmaxf(v, 0.0f);
}

// ---------------- final linear 128->40 + log_softmax ----------------
__global__ __launch_bounds__(128) void gcn_final(const float* __restrict__ h,
                                                 const float* __restrict__ lw,
                                                 const float* __restrict__ lb,
                                                 float* __restrict__ out, int n) {
    __shared__ float row[HID];
    __shared__ float lg[OUTC];
    int nd = blockIdx.x;
    int t = threadIdx.x;
    row[t] = h[(size_t)nd * HID + t];
    __syncthreads();
    if (t < OUTC) {
        float s = lb[t];
        #pragma unroll 4
        for (int k = 0; k < HID; ++k) s += row[k] * lw[k * OUTC + t];
        lg[t] = s;
    }
    __syncthreads();
    if (t < OUTC) {
        float mx = -INFINITY;
        for (int j = 0; j < OUTC; ++j) mx = fmaxf(mx, lg[j]);
        float se = 0.0f;
        for (int j = 0; j < OUTC; ++j) se += expf(lg[j] - mx);
        out[(size_t)nd * OUTC + t] = lg[t] - mx - logf(se);
    }
}

// ---------------- launcher ----------------
extern "C" void kernel_launch(void* const* d_in, const int* in_sizes, int n_in,
                              void* d_out, int out_size, void* d_ws, size_t ws_size,
                              hipStream_t stream) {
    const float* x       = (const float*)d_in[0];
    const int*   ei      = (const int*)d_in[1];    // [2, E] (JAX default x32 -> int32)
    const float* la      = (const float*)d_in[2];
    const float* lb      = (const float*)d_in[3];
    const float* mlp_w   = (const float*)d_in[4];  // [3,128,128]
    const float* mlp_b   = (const float*)d_in[5];  // [3,128]
    const float* conv_w  = (const float*)d_in[6];  // [3,128,128]
    const float* conv_b  = (const float*)d_in[7];  // [3,128]
    const float* bn_g    = (const float*)d_in[8];  // [2,128]
    const float* bn_b    = (const float*)d_in[9];  // [2,128]
    const float* lin_w   = (const float*)d_in[10]; // [128,40]
    const float* lin_b   = (const float*)d_in[11]; // [40]
    float* out = (float*)d_out;

    // workspace layout (~77 MB)
    float* ws = (float*)d_ws;
    const size_t NM = (size_t)N_NODES * HID;
    float* hA    = ws;            // current node features
    float* hB    = hA + NM;       // relu(mlp) output
    float* hC    = hB + NM;       // conv-linear output m
    float* dinv  = hC + NM;       // [N]
    float* bnsum = dinv + N_NODES;
    float* bnsq  = bnsum + HID;

    const int TPB = 256;
    const int nElemBlocks = (N_NODES * HID + TPB - 1) / TPB;
    const int gemmBlocks  = (N_NODES + 63) / 64;     // 64 rows per block (4 waves x 16)

    // 1) dinv = rsqrt(1 + indegree)
    gcn_init_deg<<<(N_NODES + TPB - 1) / TPB, TPB, 0, stream>>>(dinv, N_NODES);
    gcn_deg_edges<<<(N_EDGES + TPB - 1) / TPB, TPB, 0, stream>>>(ei, dinv, N_EDGES);
    gcn_finish_dinv<<<(N_NODES + TPB - 1) / TPB, TPB, 0, stream>>>(dinv, N_NODES);

    // 2) h0 = exp(la)*x[:, :128] + exp(lb)*x[:, 128:]
    gcn_combine<<<nElemBlocks, TPB, 0, stream>>>(x, la, lb, hA, N_NODES);

    // 3) layers
    for (int i = 0; i < NLAYERS; ++i) {
        gcn_gemm128_wmma<<<gemmBlocks, 128, 0, stream>>>(
            hA, mlp_w + (size_t)i * HID * HID, mlp_b + i * HID, hB, N_NODES, 1);
        gcn_gemm128_wmma<<<gemmBlocks, 128, 0, stream>>>(
            hB, conv_w + (size_t)i * HID * HID, nullptr, hC, N_NODES, 0);
        gcn_selfloop<<<nElemBlocks, TPB, 0, stream>>>(hC, dinv, conv_b + i * HID, hA, N_NODES);
        gcn_scatter<<<N_EDGES / 8, 256, 0, stream>>>(ei, dinv, hC, hA, N_EDGES);
        if (i < NLAYERS - 1) {
            gcn_bn_zero<<<1, 256, 0, stream>>>(bnsum, bnsq);
            gcn_bn_stats<<<256, 128, 0, stream>>>(hA, bnsum, bnsq, N_NODES);
            gcn_bn_apply<<<nElemBlocks, TPB, 0, stream>>>(
                hA, bnsum, bnsq, bn_g + i * HID, bn_b + i * HID, N_NODES);
        }
    }

    // 4) classifier + log_softmax
    gcn_final<<<N_NODES, 128, 0, stream>>>(hA, lin_w, lin_b, out, N_NODES);
}